// SGNN_72567767433498
// MI455X (gfx1250) — compile-verified
//
#include <hip/hip_runtime.h>
#include <hip/hip_bf16.h>

typedef float v2f __attribute__((ext_vector_type(2)));
typedef float v4f __attribute__((ext_vector_type(4)));
typedef float v8f __attribute__((ext_vector_type(8)));

// ---------------------------------------------------------------------------
// Zero the [N,6] node accumulator in workspace (must be re-zeroed every call).
// b128 stores; tail handled scalar.
// ---------------------------------------------------------------------------
__global__ __launch_bounds__(256) void sgnn_zero_kernel(float* __restrict__ p, int n) {
    int i = blockIdx.x * blockDim.x + threadIdx.x;
    int i4 = i * 4;
    if (i4 + 4 <= n) {
        v4f z = {0.0f, 0.0f, 0.0f, 0.0f};
        *(v4f*)(p + i4) = z;
    } else {
        for (int k = i4; k < n; ++k) p[k] = 0.0f;
    }
}

// ---------------------------------------------------------------------------
// Edge-centric stage: one thread per edge (grid-stride).
//   ea  = edge_attr * e1_fce_w.T + e1_fce_b          -> 2 vals
//   ns  = (x[src]+x[dst]) * e1_fcx_w.T + e1_fcx_b    -> 4 vals
//   edge_attr2 = relu([ea, ns])                      -> [E,6]  (output)
//   atomicAdd into node_acc[src]                     -> segment_sum scatter
// Streaming traffic (indices, attrs, edge output) uses non-temporal hints so
// x (4 MB) and node_acc (12 MB) stay resident in the 192 MB L2; gathers and
// atomics then never touch HBM.
// ---------------------------------------------------------------------------
__global__ __launch_bounds__(256) void sgnn_edge_kernel(
    const float* __restrict__ x,
    const float* __restrict__ edge_attr,
    const int*   __restrict__ src,
    const int*   __restrict__ dst,
    const float* __restrict__ e1_fcx_w,   // [4,2]
    const float* __restrict__ e1_fcx_b,   // [4]
    const float* __restrict__ e1_fce_w,   // [2,1]
    const float* __restrict__ e1_fce_b,   // [2]
    float* __restrict__ edge_out,         // [E,6]
    float* __restrict__ node_acc,         // [N,6]
    int E)
{
    __shared__ float sw[16];
    int t = threadIdx.x;
    if (t < 16) {
        float v;
        if      (t < 8)  v = e1_fcx_w[t];       // sw[0..7]  : fcx weights
        else if (t < 12) v = e1_fcx_b[t - 8];   // sw[8..11] : fcx bias
        else if (t < 14) v = e1_fce_w[t - 12];  // sw[12..13]: fce weights
        else             v = e1_fce_b[t - 14];  // sw[14..15]: fce bias
        sw[t] = v;
    }
    __syncthreads();

    float wfe0 = sw[12], wfe1 = sw[13], bfe0 = sw[14], bfe1 = sw[15];
    float w00 = sw[0], w01 = sw[1], w10 = sw[2], w11 = sw[3];
    float w20 = sw[4], w21 = sw[5], w30 = sw[6], w31 = sw[7];
    float b0 = sw[8], b1 = sw[9], b2 = sw[10], b3 = sw[11];

    int stride = gridDim.x * blockDim.x;
    for (int e = blockIdx.x * blockDim.x + t; e < E; e += stride) {
        // Prefetch next grid-stride iteration of the index streams (CDNA5
        // global_prefetch_b8; speculative — OOB silently dropped).
        __builtin_prefetch(&src[e + stride], 0, 0);
        __builtin_prefetch(&dst[e + stride], 0, 0);

        int s = __builtin_nontemporal_load(&src[e]);
        int d = __builtin_nontemporal_load(&dst[e]);
        float a = __builtin_nontemporal_load(&edge_attr[e]);

        // Gathers: regular-temporal, L2-resident.
        v2f xs = *(const v2f*)(x + (size_t)s * 2);
        v2f xd = *(const v2f*)(x + (size_t)d * 2);
        float sx = xs.x + xd.x;
        float sy = xs.y + xd.y;

        float o0 = fmaf(a, wfe0, bfe0);
        float o1 = fmaf(a, wfe1, bfe1);
        float o2 = fmaf(sx, w00, fmaf(sy, w01, b0));
        float o3 = fmaf(sx, w10, fmaf(sy, w11, b1));
        float o4 = fmaf(sx, w20, fmaf(sy, w21, b2));
        float o5 = fmaf(sx, w30, fmaf(sy, w31, b3));
        o0 = fmaxf(o0, 0.0f); o1 = fmaxf(o1, 0.0f);
        o2 = fmaxf(o2, 0.0f); o3 = fmaxf(o3, 0.0f);
        o4 = fmaxf(o4, 0.0f); o5 = fmaxf(o5, 0.0f);

        // [E,6] row store: rows are 24B, always 8B-aligned -> three NT b64 stores.
        v2f* eo = (v2f*)(edge_out + (size_t)e * 6);
        v2f p01 = {o0, o1}, p23 = {o2, o3}, p45 = {o4, o5};
        __builtin_nontemporal_store(p01, eo + 0);
        __builtin_nontemporal_store(p23, eo + 1);
        __builtin_nontemporal_store(p45, eo + 2);

        float* acc = node_acc + (size_t)s * 6;
        atomicAdd(acc + 0, o0);
        atomicAdd(acc + 1, o1);
        atomicAdd(acc + 2, o2);
        atomicAdd(acc + 3, o3);
        atomicAdd(acc + 4, o4);
        atomicAdd(acc + 5, o5);
    }
}

// ---------------------------------------------------------------------------
// Node-centric stage via V_WMMA_F32_16X16X4_F32 (full f32 precision).
// Both linears fused into one GEMM with a block-diagonal 8x16 weight matrix:
//   A[m][0..1] = x[m],  A[m][2..7] = edge_sum[m]
//   B[0..1][0..3]  = n1_fcx_w.T,  B[2..7][4..13] = n1_fce_w.T, zeros elsewhere
//   D = A @ B  ->  cols 0..3 = xn, cols 4..13 = edge_sum @ n1_fce_w.T
// K = 8 -> two chained 16x16x4 f32 WMMAs. One wave computes 16 nodes.
// ---------------------------------------------------------------------------
__global__ __launch_bounds__(256) void sgnn_node_kernel(
    const float* __restrict__ x,          // [N,2]
    const float* __restrict__ node_acc,   // [N,6]
    const float* __restrict__ n1_fcx_w,   // [4,2]
    const float* __restrict__ n1_fcx_b,   // [4]
    const float* __restrict__ n1_fce_w,   // [10,6]
    const float* __restrict__ n1_fce_b,   // [10]
    float* __restrict__ out,              // [N,14]
    int N)
{
    __shared__ float W[8][16];    // B matrix, K-major
    __shared__ float bias[16];
    int t = threadIdx.x;
    if (t < 128) {
        int k = t >> 4, n = t & 15;
        float v = 0.0f;
        if (n < 4)       { if (k < 2)  v = n1_fcx_w[n * 2 + k]; }
        else if (n < 14) { if (k >= 2) v = n1_fce_w[(n - 4) * 6 + (k - 2)]; }
        W[k][n] = v;
    } else if (t < 144) {
        int n = t - 128;
        bias[n] = (n < 4) ? n1_fcx_b[n] : (n < 14 ? n1_fce_b[n - 4] : 0.0f);
    }
    __syncthreads();

    int lane   = t & 31;
    int waveId = t >> 5;
    int half   = lane >> 4;     // 0: lanes 0-15, 1: lanes 16-31
    int mrow   = lane & 15;
    int base   = blockIdx.x * 128 + waveId * 16;
    int m      = base + mrow;
    int mc     = (m < N) ? m : (N - 1);   // clamp: EXEC must stay all-ones

    const float* xr = x        + (size_t)mc * 2;
    const float* er = node_acc + (size_t)mc * 6;

    // A layout (16x4 f32): lanes 0-15 hold K={0,1}, lanes 16-31 hold K={2,3}.
    v2f a0, a1;
    if (half == 0) { a0.x = xr[0]; a0.y = xr[1]; a1.x = er[2]; a1.y = er[3]; }
    else           { a0.x = er[0]; a0.y = er[1]; a1.x = er[4]; a1.y = er[5]; }

    // B layout (4x16 f32): VGPR0 = rows {0,2}, VGPR1 = rows {1,3} by lane half.
    int kb = half * 2;
    v2f bm0, bm1;
    bm0.x = W[kb + 0][mrow]; bm0.y = W[kb + 1][mrow];
    bm1.x = W[kb + 4][mrow]; bm1.y = W[kb + 5][mrow];

    v8f c = {};
    c = __builtin_amdgcn_wmma_f32_16x16x4_f32(false, a0, false, bm0,
                                              (short)0, c, false, false);
    c = __builtin_amdgcn_wmma_f32_16x16x4_f32(false, a1, false, bm1,
                                              (short)0, c, false, false);

    // Bias + relu in VALU.
    float bn = bias[mrow];
    float vals[8];
    #pragma unroll
    for (int r = 0; r < 8; ++r) vals[r] = fmaxf(c[r] + bn, 0.0f);

    // D layout: VGPR r -> row (base + r + 8*half), col = lane%16.
    // Fast path (all rows in range): single divergence mask (col < 14),
    // one base address, 8 NT stores at immediate offsets (row stride = 56 B).
    float* po = out + (size_t)(base + half * 8) * 14 + mrow;
    if (base + 16 <= N) {
        if (mrow < 14) {
            #pragma unroll
            for (int r = 0; r < 8; ++r)
                __builtin_nontemporal_store(vals[r], po + (size_t)r * 14);
        }
    } else {
        #pragma unroll
        for (int r = 0; r < 8; ++r) {
            int row = base + r + half * 8;
            if (row < N && mrow < 14)
                __builtin_nontemporal_store(vals[r], po + (size_t)r * 14);
        }
    }
}

// ---------------------------------------------------------------------------
extern "C" void kernel_launch(void* const* d_in, const int* in_sizes, int n_in,
                              void* d_out, int out_size, void* d_ws, size_t ws_size,
                              hipStream_t stream) {
    const float* x         = (const float*)d_in[0];
    const float* edge_attr = (const float*)d_in[1];
    const int*   edge_idx  = (const int*)  d_in[2];
    const float* e1_fcx_w  = (const float*)d_in[3];
    const float* e1_fcx_b  = (const float*)d_in[4];
    const float* e1_fce_w  = (const float*)d_in[5];
    const float* e1_fce_b  = (const float*)d_in[6];
    const float* n1_fcx_w  = (const float*)d_in[7];
    const float* n1_fcx_b  = (const float*)d_in[8];
    const float* n1_fce_w  = (const float*)d_in[9];
    const float* n1_fce_b  = (const float*)d_in[10];

    int N = in_sizes[0] / 2;
    int E = in_sizes[1];
    const int* src = edge_idx;
    const int* dst = edge_idx + E;

    float* x2_out   = (float*)d_out;                    // [N,14]
    float* edge_out = (float*)d_out + (size_t)N * 14;   // [E,6]
    float* node_acc = (float*)d_ws;                     // [N,6] scratch

    int zn = N * 6;
    int zthreads = (zn + 3) / 4;
    sgnn_zero_kernel<<<(zthreads + 255) / 256, 256, 0, stream>>>(node_acc, zn);

    // Grid-stride with a capped grid so the index-stream prefetch distance is
    // a real next-iteration address (~5 edges/thread at E = 10M).
    int eblocks = (E + 255) / 256;
    if (eblocks > 8192) eblocks = 8192;
    sgnn_edge_kernel<<<eblocks, 256, 0, stream>>>(
        x, edge_attr, src, dst,
        e1_fcx_w, e1_fcx_b, e1_fce_w, e1_fce_b,
        edge_out, node_acc, E);

    sgnn_node_kernel<<<(N + 127) / 128, 256, 0, stream>>>(
        x, node_acc, n1_fcx_w, n1_fcx_b, n1_fce_w, n1_fce_b, x2_out, N);
}